// SatFormer_Block_20383914787570
// MI455X (gfx1250) — compile-verified
//
#include <hip/hip_runtime.h>
#include <hip/hip_bf16.h>

typedef float v2f __attribute__((ext_vector_type(2)));
typedef float v8f __attribute__((ext_vector_type(8)));

// Sum across each 16-lane half of the wave (wave32). Result uniform per half.
__device__ __forceinline__ float hsum16(float v) {
    v += __shfl_xor(v, 1, 32);
    v += __shfl_xor(v, 2, 32);
    v += __shfl_xor(v, 4, 32);
    v += __shfl_xor(v, 8, 32);
    return v;
}

__global__ void zero_ws_kernel(float* ws) {
    if (threadIdx.x < 32) ws[threadIdx.x] = 0.f;
}

// Column sums of x [S,16] -> xsum[16]
__global__ void colsum_kernel(const float* __restrict__ x, float* __restrict__ xsum, int S) {
    __shared__ float sm[256];
    const int tid = threadIdx.x;
    const int c = tid & 15, ty = tid >> 4;
    const int rowsPerBlock = S / gridDim.x;
    const int base = blockIdx.x * rowsPerBlock;
    float s = 0.f;
    for (int r = base + ty; r < base + rowsPerBlock; r += 16)
        s += x[(size_t)r * 16 + c];
    sm[tid] = s;
    __syncthreads();
    if (tid < 16) {
        float t = 0.f;
        #pragma unroll
        for (int j = 0; j < 16; ++j) t += sm[j * 16 + tid];
        atomicAdd(&xsum[tid], t);
    }
}

// One wave per 16-row tile. Q/K projections via V_WMMA_F32_16X16X4_F32 (x4 over K).
__global__ __launch_bounds__(256) void satformer_main(
    const float* __restrict__ x,
    const float* __restrict__ Wq, const float* __restrict__ bq,
    const float* __restrict__ Wk, const float* __restrict__ bk,
    const float* __restrict__ Wv, const float* __restrict__ bv,
    const int* __restrict__ headsp,
    const float* __restrict__ xsum, float* __restrict__ bucket, int S)
{
    const int lane = threadIdx.x & 31;
    const int wave = threadIdx.x >> 5;
    const int li = lane & 15;        // column / feature index within half
    const int h  = lane >> 4;        // half: 0 -> rows 0..7, 1 -> rows 8..15 (D layout)
    const int tile = blockIdx.x * 8 + wave;
    const int r0 = tile * 16;
    if (r0 >= S) return;             // wave-uniform; EXEC stays all-1s for WMMA

    const int heads = headsp[0];
    const float scale = rsqrtf((float)(16 / heads));   // head_dim^-0.5

    const float bq_l = bq[li];
    const float bk_l = bk[li];

    // colsum(Wv)[li] and sum(bv); VsumTotal = xsum . colsum(Wv) + S*sum(bv)
    float wvs = 0.f;
    #pragma unroll
    for (int d = 0; d < 16; ++d) wvs += Wv[d * 16 + li];
    float bvs = 0.f;
    #pragma unroll
    for (int d = 0; d < 16; ++d) bvs += bv[d];
    const float Vtot = hsum16(xsum[li] * wvs) + (float)S * bvs;

    // ---- WMMA projections: Q = X*Wq^T + bq, K = X*Wk^T + bk -------------
    // A (16x4 f32): lane holds X[r0+li, 4t+2h .. +1];  B (4x16): B[k,n]=W[n,k]
    const float* xrow = x  + (size_t)(r0 + li) * 16;
    const float* wqr  = Wq + (size_t)li * 16;
    const float* wkr  = Wk + (size_t)li * 16;
    v8f cq = {0.f,0.f,0.f,0.f,0.f,0.f,0.f,0.f};
    v8f ck = {0.f,0.f,0.f,0.f,0.f,0.f,0.f,0.f};
    #pragma unroll
    for (int t = 0; t < 4; ++t) {
        const int off = 4 * t + 2 * h;
        v2f a   = *(const v2f*)(xrow + off);
        v2f bqv = *(const v2f*)(wqr  + off);
        v2f bkv = *(const v2f*)(wkr  + off);
        cq = __builtin_amdgcn_wmma_f32_16x16x4_f32(false, a, false, bqv, (short)0, cq, false, false);
        ck = __builtin_amdgcn_wmma_f32_16x16x4_f32(false, a, false, bkv, (short)0, ck, false, false);
    }
    #pragma unroll
    for (int r = 0; r < 8; ++r) { cq[r] += bq_l; ck[r] += bk_l; }
    // D layout now: lane(h,li), VGPR r <-> Q[r+8h, li], K[r+8h, li]

    // ---- tile-boundary neighbors (rows r0-1 and r0+16) -------------------
    float s_prev_b = 0.f, vr_prev_b = 0.f;
    if (r0 > 0) {
        const float* xp = x + (size_t)(r0 - 1) * 16;
        float kp = 0.f;
        #pragma unroll
        for (int k2 = 0; k2 < 16; ++k2) kp += xp[k2] * wkr[k2];
        kp += bk_l;                                   // K[prev, d=li]
        s_prev_b  = __shfl(hsum16(cq[0] * kp), 0, 32);    // s(r0, r0-1) from lo half
        vr_prev_b = hsum16(xp[li] * wvs) + bvs;
    }
    float s_next_b = 0.f, vr_next_b = 0.f;
    if (r0 + 16 < S) {
        const float* xn = x + (size_t)(r0 + 16) * 16;
        float kn = 0.f;
        #pragma unroll
        for (int k2 = 0; k2 < 16; ++k2) kn += xn[k2] * wkr[k2];
        kn += bk_l;
        s_next_b  = __shfl(hsum16(cq[7] * kn), 16, 32);   // s(r0+15, r0+16) from hi half
        vr_next_b = hsum16(xn[li] * wvs) + bvs;
    }

    // ---- tridiagonal scores + row-sums of V, per row -----------------------
    float s_diag = 0.f, s_sup = 0.f, s_sub = 0.f, my_vr = 0.f;
    #pragma unroll
    for (int v = 0; v < 8; ++v) {
        const float qv = cq[v];
        const float dval = hsum16(qv * ck[v]);                      // s(i,i), i=v+8h
        const float ksup = (v < 7) ? ck[v + 1] : __shfl_xor(ck[0], 16, 32); // row 8 -> lo half
        const float sval = hsum16(qv * ksup);                       // s(i,i+1)
        const float ksub = (v > 0) ? ck[v - 1] : __shfl_xor(ck[7], 16, 32); // row 7 -> hi half
        const float bval = hsum16(qv * ksub);                       // s(i,i-1)
        const float xv   = x[(size_t)(r0 + v + 8 * h) * 16 + li];
        const float vval = hsum16(xv * wvs) + bvs;                  // vr_i
        if (li == v) { s_diag = dval; s_sup = sval; s_sub = bval; my_vr = vval; }
    }
    // row lanes: row i<8 -> lane i; row i>=8 -> lane i+8
    const int my_row = li + 8 * h;
    if (my_row == 0)  s_sub = s_prev_b;   // cross-tile fixups
    if (my_row == 15) s_sup = s_next_b;

    const int pr  = my_row > 0 ? my_row - 1 : 0;
    const int nx  = my_row < 15 ? my_row + 1 : 15;
    const int prl = pr < 8 ? pr : pr + 8;
    const int nxl = nx < 8 ? nx : nx + 8;
    float vrp = __shfl(my_vr, prl, 32);
    float vrn = __shfl(my_vr, nxl, 32);
    if (my_row == 0)  vrp = vr_prev_b;
    if (my_row == 15) vrn = vr_next_b;

    const bool has_sub = (r0 + my_row) > 0;
    const bool has_sup = (r0 + my_row) < (S - 1);
    const float wsub = has_sub ? (expf(scale * s_sub) - 1.f) : 0.f;
    const float wdia = expf(scale * s_diag) - 1.f;
    const float wsup = has_sup ? (expf(scale * s_sup) - 1.f) : 0.f;

    const float denom = (float)S + wsub + wdia + wsup;
    const float numer = Vtot + wsub * vrp + wdia * my_vr + wsup * vrn;
    // bucket index = (r0+my_row) % 16 == my_row (r0 multiple of 16)
    if (li < 8) atomicAdd(&bucket[my_row], numer / denom);
}

__global__ void finalize_kernel(const float* __restrict__ bucket, float* __restrict__ out, int S) {
    const int j = threadIdx.x;
    if (j < 128) out[j] = bucket[j >> 3] * (1.f / (float)S);
}

extern "C" void kernel_launch(void* const* d_in, const int* in_sizes, int n_in,
                              void* d_out, int out_size, void* d_ws, size_t ws_size,
                              hipStream_t stream) {
    const float* x  = (const float*)d_in[0];
    const float* Wq = (const float*)d_in[1];
    const float* bq = (const float*)d_in[2];
    const float* Wk = (const float*)d_in[3];
    const float* bk = (const float*)d_in[4];
    const float* Wv = (const float*)d_in[5];
    const float* bv = (const float*)d_in[6];
    const int* heads = (const int*)d_in[7];
    float* out = (float*)d_out;

    float* wsf    = (float*)d_ws;
    float* xsum   = wsf;        // 16 floats
    float* bucket = wsf + 16;   // 16 floats

    const int S = in_sizes[0] / 16;

    zero_ws_kernel<<<1, 32, 0, stream>>>(wsf);
    colsum_kernel<<<64, 256, 0, stream>>>(x, xsum, S);
    const int tiles  = S / 16;
    const int blocks = (tiles + 7) / 8;
    satformer_main<<<blocks, 256, 0, stream>>>(x, Wq, bq, Wk, bk, Wv, bv, heads, xsum, bucket, S);
    finalize_kernel<<<1, 128, 0, stream>>>(bucket, out, S);
}